// ResidualGNN_39247411151345
// MI455X (gfx1250) — compile-verified
//
#include <hip/hip_runtime.h>
#include <cmath>

#define F 128  // feature width of both GCN layers

typedef float v2f __attribute__((ext_vector_type(2)));
typedef float v8f __attribute__((ext_vector_type(8)));

// ---------------------------------------------------------------------------
// Degree / normalization:  deg[i] = 1 (self loop) + #incoming edges,
// then in-place dinv[i] = rsqrt(deg[i]).
// ---------------------------------------------------------------------------
__global__ void __launch_bounds__(256)
init_deg_kernel(float* __restrict__ deg, int N) {
  int i = blockIdx.x * blockDim.x + threadIdx.x;
  if (i < N) deg[i] = 1.0f;
}

__global__ void __launch_bounds__(256)
accum_deg_kernel(const long long* __restrict__ dst, float* __restrict__ deg, int E) {
  int e = blockIdx.x * blockDim.x + threadIdx.x;
  if (e < E) unsafeAtomicAdd(&deg[(int)dst[e]], 1.0f);
}

__global__ void __launch_bounds__(256)
dinv_kernel(float* __restrict__ deg, int N) {
  int i = blockIdx.x * blockDim.x + threadIdx.x;
  if (i < N) deg[i] = rsqrtf(deg[i]);
}

// ---------------------------------------------------------------------------
// Hs = rowscale(dinv) * (X @ W), and S = Hs (self-loop init of the
// aggregation accumulator).  Full-fp32 WMMA: V_WMMA_F32_16X16X4_F32.
// 256 threads = 8 waves per block; each wave owns 16 output rows x 128 cols
// (8 accumulators of 16x16), so one 64KB LDS staging of W^T is shared by
// 8 waves (~4 blocks / 32 waves per WGP within the 320KB LDS pool).
// X may alias Hs (GEMM is row-local), used for the in-place layer-2 pass.
// ---------------------------------------------------------------------------
__global__ void __launch_bounds__(256)
gcn_gemm_scale_kernel(const float* X,                   // [N,128] (may alias Hs)
                      const float* __restrict__ W,      // [128,128] row-major
                      const float* __restrict__ dinv,   // [N]
                      float* Hs,                        // [N,128]
                      float* __restrict__ S,            // [N,128]
                      int N) {
  __shared__ float Wt[F * F];  // Wt[n*128 + k] = W[k*128 + n]
  for (int idx = threadIdx.x; idx < F * F; idx += 256) {
    int k = idx >> 7, n = idx & 127;
    Wt[n * F + k] = W[idx];
  }
  __syncthreads();

  const int lane = threadIdx.x & 31;
  const int wave = threadIdx.x >> 5;
  const int m    = lane & 15;   // row (A) / col (B,D) within 16x16 tile
  const int hi   = lane >> 4;   // 0 -> K{0,1}, 1 -> K{2,3} per ISA A layout
  const int rowBase = blockIdx.x * 128 + wave * 16;

  const int arow = min(rowBase + m, N - 1);
  const float* xrow = X + (size_t)arow * F + hi * 2;

  v8f acc[8] = {};
#pragma unroll 4
  for (int ks = 0; ks < F / 4; ++ks) {
    v2f a = *(const v2f*)(xrow + ks * 4);  // A frag: 16x4 f32, 2 VGPRs
#pragma unroll
    for (int nt = 0; nt < 8; ++nt) {
      // B frag: 4x16 f32; lane m = col, K pair contiguous in Wt
      v2f b = *(const v2f*)(&Wt[(nt * 16 + m) * F + ks * 4 + hi * 2]);
      acc[nt] = __builtin_amdgcn_wmma_f32_16x16x4_f32(
          false, a, false, b, (short)0, acc[nt], false, false);
    }
  }

  // D layout: VGPR v -> row rowBase+v (lanes 0-15) / rowBase+8+v (lanes 16-31)
  const int roff = rowBase + hi * 8;
  float dv[8];
#pragma unroll
  for (int v = 0; v < 8; ++v) dv[v] = dinv[min(roff + v, N - 1)];

#pragma unroll
  for (int nt = 0; nt < 8; ++nt) {
#pragma unroll
    for (int v = 0; v < 8; ++v) {
      int r = roff + v;
      if (r < N) {
        float val = acc[nt][v] * dv[v];
        size_t a2 = (size_t)r * F + nt * 16 + m;
        Hs[a2] = val;
        S[a2]  = val;
      }
    }
  }
}

// ---------------------------------------------------------------------------
// Edge scatter: S[dst] += Hs[src].  One wave per edge; each lane moves a
// float4 and issues 4 global_atomic_add_f32 (h fits in the 192MB L2, so
// gathers and atomics both resolve at L2).
// ---------------------------------------------------------------------------
__global__ void __launch_bounds__(256)
scatter_kernel(const long long* __restrict__ src, const long long* __restrict__ dst,
               const float* __restrict__ Hs, float* __restrict__ S, int E) {
  int lane = threadIdx.x & 31;
  int edge = blockIdx.x * 8 + (threadIdx.x >> 5);
  if (edge >= E) return;
  int s = (int)src[edge];
  int d = (int)dst[edge];
  const float4 v = *(const float4*)(Hs + (size_t)s * F + lane * 4);
  float* p = S + (size_t)d * F + lane * 4;
  unsafeAtomicAdd(p + 0, v.x);
  unsafeAtomicAdd(p + 1, v.y);
  unsafeAtomicAdd(p + 2, v.z);
  unsafeAtomicAdd(p + 3, v.w);
}

// ---------------------------------------------------------------------------
// out = relu(dinv[row] * S + bias[col]); float4-vectorized, in-place safe.
// ---------------------------------------------------------------------------
__global__ void __launch_bounds__(256)
finalize_kernel(const float4* __restrict__ S, const float* __restrict__ dinv,
                const float4* __restrict__ bias, float4* __restrict__ out, int N) {
  int idx = blockIdx.x * blockDim.x + threadIdx.x;  // over N * 32 float4s
  if (idx < N * (F / 4)) {
    int i  = idx >> 5;          // node
    int f4 = idx & 31;          // float4 within row
    float d = dinv[i];
    float4 s = S[idx];
    float4 b = bias[f4];
    float4 r;
    r.x = fmaf(d, s.x, b.x); r.x = r.x > 0.0f ? r.x : 0.0f;
    r.y = fmaf(d, s.y, b.y); r.y = r.y > 0.0f ? r.y : 0.0f;
    r.z = fmaf(d, s.z, b.z); r.z = r.z > 0.0f ? r.z : 0.0f;
    r.w = fmaf(d, s.w, b.w); r.w = r.w > 0.0f ? r.w : 0.0f;
    out[idx] = r;
  }
}

// ---------------------------------------------------------------------------
// Classifier: out[i,c] = tanh(h[i,:] @ Wfc[:,c] + bfc[c]),  c = 0..2.
// Wfc/bfc indices are wave-uniform -> compiler can use SMEM broadcasts.
// ---------------------------------------------------------------------------
__global__ void __launch_bounds__(256)
fc_tanh_kernel(const float* __restrict__ H, const float* __restrict__ Wfc,
               const float* __restrict__ bfc, float* __restrict__ out, int N) {
  int i = blockIdx.x * blockDim.x + threadIdx.x;
  if (i >= N) return;
  float a0 = bfc[0], a1 = bfc[1], a2 = bfc[2];
  const float4* h = (const float4*)(H + (size_t)i * F);
#pragma unroll 4
  for (int f4 = 0; f4 < F / 4; ++f4) {
    float4 x = h[f4];
    int f = f4 * 4;
    a0 = fmaf(x.x, Wfc[(f + 0) * 3 + 0], a0);
    a1 = fmaf(x.x, Wfc[(f + 0) * 3 + 1], a1);
    a2 = fmaf(x.x, Wfc[(f + 0) * 3 + 2], a2);
    a0 = fmaf(x.y, Wfc[(f + 1) * 3 + 0], a0);
    a1 = fmaf(x.y, Wfc[(f + 1) * 3 + 1], a1);
    a2 = fmaf(x.y, Wfc[(f + 1) * 3 + 2], a2);
    a0 = fmaf(x.z, Wfc[(f + 2) * 3 + 0], a0);
    a1 = fmaf(x.z, Wfc[(f + 2) * 3 + 1], a1);
    a2 = fmaf(x.z, Wfc[(f + 2) * 3 + 2], a2);
    a0 = fmaf(x.w, Wfc[(f + 3) * 3 + 0], a0);
    a1 = fmaf(x.w, Wfc[(f + 3) * 3 + 1], a1);
    a2 = fmaf(x.w, Wfc[(f + 3) * 3 + 2], a2);
  }
  out[(size_t)i * 3 + 0] = tanhf(a0);
  out[(size_t)i * 3 + 1] = tanhf(a1);
  out[(size_t)i * 3 + 2] = tanhf(a2);
}

// ---------------------------------------------------------------------------
extern "C" void kernel_launch(void* const* d_in, const int* in_sizes, int n_in,
                              void* d_out, int out_size, void* d_ws, size_t ws_size,
                              hipStream_t stream) {
  const float* x        = (const float*)d_in[0];
  const long long* eidx = (const long long*)d_in[1];  // [2,E] int64
  const float* W1  = (const float*)d_in[2];
  const float* b1  = (const float*)d_in[3];
  const float* W2  = (const float*)d_in[4];
  const float* b2  = (const float*)d_in[5];
  const float* Wfc = (const float*)d_in[6];
  const float* bfc = (const float*)d_in[7];
  float* out = (float*)d_out;

  const int N = in_sizes[0] / F;
  const int E = in_sizes[1] / 2;
  const long long* src = eidx;
  const long long* dst = eidx + E;

  // workspace: dinv [N] | bufA [N*F] | bufB [N*F]
  char* ws = (char*)d_ws;
  float* dinv = (float*)ws;
  size_t off = (((size_t)N * 4) + 255) & ~(size_t)255;
  float* bufA = (float*)(ws + off);
  float* bufB = bufA + (size_t)N * F;

  const int T = 256;
  // degrees + dinv
  init_deg_kernel<<<(N + T - 1) / T, T, 0, stream>>>(dinv, N);
  accum_deg_kernel<<<(E + T - 1) / T, T, 0, stream>>>(dst, dinv, E);
  dinv_kernel<<<(N + T - 1) / T, T, 0, stream>>>(dinv, N);

  const int gemmGrid = (N + 127) / 128;
  const int scatGrid = (E + 7) / 8;
  const int vecGrid  = (N * (F / 4) + T - 1) / T;

  // ---- layer 1: Hs=bufA, S=bufB -> h1 in bufB
  gcn_gemm_scale_kernel<<<gemmGrid, 256, 0, stream>>>(x, W1, dinv, bufA, bufB, N);
  scatter_kernel<<<scatGrid, T, 0, stream>>>(src, dst, bufA, bufB, E);
  finalize_kernel<<<vecGrid, T, 0, stream>>>((const float4*)bufB, dinv,
                                             (const float4*)b1, (float4*)bufB, N);

  // ---- layer 2 (GEMM in place on bufB): Hs=bufB, S=bufA -> h2 in bufA
  gcn_gemm_scale_kernel<<<gemmGrid, 256, 0, stream>>>(bufB, W2, dinv, bufB, bufA, N);
  scatter_kernel<<<scatGrid, T, 0, stream>>>(src, dst, bufB, bufA, E);
  finalize_kernel<<<vecGrid, T, 0, stream>>>((const float4*)bufA, dinv,
                                             (const float4*)b2, (float4*)bufA, N);

  // ---- classifier
  fc_tanh_kernel<<<(N + T - 1) / T, T, 0, stream>>>(bufA, Wfc, bfc, out, N);
}